// VectorQuantizerEMA_7576322310659
// MI455X (gfx1250) — compile-verified
//
#include <hip/hip_runtime.h>
#include <hip/hip_bf16.h>
#include <math.h>

typedef __attribute__((ext_vector_type(2))) float v2f;
typedef __attribute__((ext_vector_type(8))) float v8f;
typedef __attribute__((ext_vector_type(4))) unsigned int u32x4;
typedef __attribute__((ext_vector_type(8))) int i32x8;
typedef __attribute__((ext_vector_type(4))) int i32x4;

#define N_ROWS        65536      // 64*1024 flattened rows
#define C_DIM         64
#define K_CODES       1024
#define LDS_STRIDE    68         // padded codebook row stride (floats): 64 data + 4 pad
#define ENORM_OFF     (K_CODES * LDS_STRIDE)            // e-norm table after codebook
#define SMEM_FLOATS   (ENORM_OFF + K_CODES)
#define SMEM_BYTES    (SMEM_FLOATS * 4)                 // 282624 B < 320KB WGP LDS
#define ROWS_PER_BLOCK 128       // 8 waves * 16 rows

// d_out layout (floats): loss | quantized_st | perplexity | indices | avg_probs
#define OFF_LOSS  0
#define OFF_Q     1
#define OFF_PERP  (1 + N_ROWS * C_DIM)
#define OFF_IDX   (OFF_PERP + 1)
#define OFF_AVG   (OFF_IDX + N_ROWS)

// d_ws layout (floats): e_norms[1024] | hist[1024] (as uint) | sse[1]
#define WS_ENORM  0
#define WS_HIST   1024
#define WS_SSE    2048

// ---------------------------------------------------------------- e-norms
__global__ void vq_enorm_kernel(const float* __restrict__ emb,
                                float* __restrict__ ws) {
    int k = blockIdx.x * blockDim.x + threadIdx.x;   // 0..1023
    const float4* e = (const float4*)(emb + k * C_DIM);
    float s = 0.f;
#pragma unroll
    for (int i = 0; i < C_DIM / 4; ++i) {
        float4 v = e[i];
        s += v.x * v.x + v.y * v.y + v.z * v.z + v.w * v.w;
    }
    ws[WS_ENORM + k] = s;
}

// ---------------------------------------------------------------- main
__global__ void __launch_bounds__(256)
vq_main_kernel(const float* __restrict__ in, const float* __restrict__ emb,
               float* __restrict__ out, float* __restrict__ ws) {
    extern __shared__ float smem[];                  // codebook (padded) | e-norms
    unsigned* hist = (unsigned*)(ws + WS_HIST);

    const int lane = threadIdx.x & 31;
    const int wave = threadIdx.x >> 5;

    // ---- stage codebook into LDS with 68-dword padded rows
#if defined(__HIP_DEVICE_COMPILE__) && __has_builtin(__builtin_amdgcn_tensor_load_to_lds)
    if (wave == 0) {
        // Tensor Data Mover: 2D 1024x64 f32 tile, HW-inserted 4-dword row pad.
        unsigned long long ga = (unsigned long long)emb;
        u32x4 g0;
        g0.x = 1u;                                   // count=1, no gather, user D#
        g0.y = 0u;                                   // lds_addr = 0 (base of dyn LDS)
        g0.z = (unsigned int)(ga & 0xffffffffull);   // global_addr[31:0]
        g0.w = (unsigned int)((ga >> 32) & 0x1ffffffull) | (2u << 30); // addr[56:32] | type=2
        i32x8 g1;
        g1[0] = (2 << 16)                            // data_size = 4 bytes
              | (1 << 20)                            // pad_enable
              | (5 << 22)                            // pad_interval: 64 dwords
              | (3 << 25);                           // pad_amount: 4 dwords
        g1[1] = (C_DIM & 0xffff) << 16;              // tensor_dim0 = 64   [63:48]
        g1[2] = (K_CODES & 0xffff) << 16;            // tensor_dim1 = 1024 [95:80]
        g1[3] = (C_DIM & 0xffff) << 16;              // tile_dim0 = 64     [127:112]
        g1[4] = K_CODES;                             // tile_dim1 = 1024   [143:128]
        g1[5] = C_DIM;                               // tensor_dim0_stride = 64
        g1[6] = 0;
        g1[7] = 0;
        i32x4 gz4 = {0, 0, 0, 0};
        i32x8 gz8 = {0, 0, 0, 0, 0, 0, 0, 0};
        __builtin_amdgcn_tensor_load_to_lds(g0, g1, gz4, gz4, gz8, 0);
    }
    // e-norm table -> LDS (independent region) while TDM streams the codebook
    {
        int k = threadIdx.x;
#pragma unroll
        for (int u = 0; u < K_CODES / 256; ++u, k += 256)
            smem[ENORM_OFF + k] = ws[WS_ENORM + k];
    }
    if (wave == 0) __builtin_amdgcn_s_wait_tensorcnt(0);
    __syncthreads();
#else
    {
        const float4* esrc = (const float4*)emb;
#pragma unroll 4
        for (int u = 0; u < (K_CODES * C_DIM / 4) / 256; ++u) {
            int l4  = threadIdx.x + u * 256;
            int row = l4 >> 4;
            int c4  = l4 & 15;
            float4 v = esrc[l4];
            *(float4*)&smem[row * LDS_STRIDE + c4 * 4] = v;
        }
        int k = threadIdx.x;
        for (int u = 0; u < K_CODES / 256; ++u, k += 256)
            smem[ENORM_OFF + k] = ws[WS_ENORM + k];
        __syncthreads();
    }
#endif

    const int rowbase = blockIdx.x * ROWS_PER_BLOCK + wave * 16;
    const int myrow   = rowbase + (lane & 15);
    const int koff    = (lane >> 4) * 2;             // ISA A-layout: halves hold k0/k0+1 vs k0+2/k0+3
    const int mycol   = lane & 15;

    // ---- A operands: 16 k-chunks of the 16x64 row tile, WMMA 16x16x4 layout
    v2f a[16];
#pragma unroll
    for (int kc = 0; kc < 16; ++kc)
        a[kc] = *(const v2f*)(in + myrow * C_DIM + kc * 4 + koff);

    float bestv[8];
    int   bestn[8];
#pragma unroll
    for (int j = 0; j < 8; ++j) { bestv[j] = __FLT_MAX__; bestn[j] = 0; }

    // ---- sweep K=1024 codes in 16-column tiles; unroll 2 => two independent
    //      WMMA accumulation chains to hide XDL latency; all operands from LDS
#pragma unroll 2
    for (int nt = 0; nt < K_CODES / 16; ++nt) {
        const int   n  = nt * 16 + mycol;            // this lane's codebook column
        const float en = smem[ENORM_OFF + n];
        v8f c = {};
#pragma unroll
        for (int kc = 0; kc < 16; ++kc) {
            v2f b = *(const v2f*)&smem[n * LDS_STRIDE + kc * 4 + koff];
            c = __builtin_amdgcn_wmma_f32_16x16x4_f32(
                    false, a[kc], false, b, (short)0, c, false, false);
        }
#pragma unroll
        for (int j = 0; j < 8; ++j) {                // dist = ||e||^2 - 2*dot (||x||^2 row-const)
            float d = en - 2.0f * c[j];
            if (d < bestv[j]) { bestv[j] = d; bestn[j] = n; }
        }
    }

    // ---- argmin across the 16 lanes of each half (C/D layout: lane==column)
#pragma unroll
    for (int j = 0; j < 8; ++j) {
#pragma unroll
        for (int off = 1; off < 16; off <<= 1) {
            float ov = __shfl_xor(bestv[j], off, 32);
            int   on = __shfl_xor(bestn[j], off, 32);
            if (ov < bestv[j] || (ov == bestv[j] && on < bestn[j])) {
                bestv[j] = ov; bestn[j] = on;
            }
        }
    }
    // broadcast winning index per row to all 32 lanes (rows 0-7 in lanes 0-15, 8-15 in 16-31)
    int idx_row[16];
#pragma unroll
    for (int r = 0; r < 16; ++r)
        idx_row[r] = __shfl(bestn[r & 7], (r >= 8) ? 16 : 0, 32);

    // ---- gather quantized rows from LDS, write outputs, accumulate SSE + histogram
    float sse = 0.f;
#pragma unroll
    for (int r = 0; r < 16; ++r) {
        const int bi   = idx_row[r];
        const int grow = rowbase + r;
        v2f q = *(const v2f*)&smem[bi * LDS_STRIDE + lane * 2];
        v2f x = *(const v2f*)(in + grow * C_DIM + lane * 2);
        float dx = q.x - x.x, dy = q.y - x.y;
        sse += dx * dx + dy * dy;
        *(v2f*)(out + OFF_Q + grow * C_DIM + lane * 2) = q;   // quantized_st == quantized (fwd)
        if (lane == 0) {
            out[OFF_IDX + grow] = (float)bi;
            atomicAdd(&hist[bi], 1u);
        }
    }
#pragma unroll
    for (int off = 16; off >= 1; off >>= 1) sse += __shfl_xor(sse, off, 32);
    if (lane == 0) atomicAdd(&ws[WS_SSE], sse);
}

// ---------------------------------------------------------------- finalize
__global__ void __launch_bounds__(1024)
vq_finalize_kernel(const unsigned* __restrict__ hist,
                   const float* __restrict__ sse,
                   float* __restrict__ out) {
    __shared__ float part[32];
    int k = threadIdx.x;                             // 0..1023
    float p = (float)hist[k] * (1.0f / (float)N_ROWS);
    out[OFF_AVG + k] = p;
    float t = p * logf(p + 1e-10f);
#pragma unroll
    for (int off = 16; off >= 1; off >>= 1) t += __shfl_xor(t, off, 32);
    if ((threadIdx.x & 31) == 0) part[threadIdx.x >> 5] = t;
    __syncthreads();
    if (threadIdx.x < 32) {
        float s = part[threadIdx.x];
#pragma unroll
        for (int off = 16; off >= 1; off >>= 1) s += __shfl_xor(s, off, 32);
        if (threadIdx.x == 0) {
            out[OFF_PERP] = expf(-s);
            out[OFF_LOSS] = 0.25f * sse[0] / ((float)N_ROWS * (float)C_DIM);
        }
    }
}

// ---------------------------------------------------------------- launch
extern "C" void kernel_launch(void* const* d_in, const int* in_sizes, int n_in,
                              void* d_out, int out_size, void* d_ws, size_t ws_size,
                              hipStream_t stream) {
    const float* in  = (const float*)d_in[0];        // [64,1024,64]
    const float* emb = (const float*)d_in[1];        // [1024,64]
    float* out = (float*)d_out;
    float* ws  = (float*)d_ws;

    // allow >64KB dynamic LDS (CDNA5 WGP has 320KB)
    (void)hipFuncSetAttribute((const void*)vq_main_kernel,
                              hipFuncAttributeMaxDynamicSharedMemorySize, SMEM_BYTES);

    // zero histogram + SSE accumulator each call (harness does not re-poison)
    (void)hipMemsetAsync((char*)d_ws + WS_HIST * sizeof(float), 0,
                         (K_CODES + 1) * sizeof(float), stream);

    vq_enorm_kernel<<<K_CODES / 256, 256, 0, stream>>>(emb, ws);
    vq_main_kernel<<<N_ROWS / ROWS_PER_BLOCK, 256, SMEM_BYTES, stream>>>(in, emb, out, ws);
    vq_finalize_kernel<<<1, K_CODES, 0, stream>>>((const unsigned*)(ws + WS_HIST),
                                                  ws + WS_SSE, out);
}